// ResGCNLayer_30210799960859
// MI455X (gfx1250) — compile-verified
//
#include <hip/hip_runtime.h>
#include <hip/hip_bf16.h>
#include <stddef.h>

#define N_NODES   50000
#define N_EDGES   800000
#define D         128
#define ROW_BLOCKS (N_NODES / 16)      // 3125, exact
#define BN_EPS    1e-5f

typedef float v2f __attribute__((ext_vector_type(2)));
typedef float v8f __attribute__((ext_vector_type(8)));

// ---- workspace layout (float offsets) ----
#define DEG_OUT_OFF 0
#define DEG_IN_OFF  (N_NODES)
#define COLSUM_OFF  (2 * N_NODES)
#define COLSQ_OFF   (2 * N_NODES + 128)
#define AB_OFF      (2 * N_NODES + 256)          // [0..127]=scale a, [128..255]=shift c
#define AGG_OFF     (2 * N_NODES + 512)          // 16B aligned
#define ZERO_FLOATS (AGG_OFF + N_NODES * D)      // 6,500,512 (div by 4)

// -------------------------------------------------------------------------
// 0) zero the workspace (deg, colsums, agg) — agg must be zero before scatter
__global__ void zero_kernel(float4* __restrict__ p, int n4) {
    for (int i = blockIdx.x * blockDim.x + threadIdx.x; i < n4;
         i += gridDim.x * blockDim.x) {
        p[i] = make_float4(0.f, 0.f, 0.f, 0.f);
    }
}

// 1) degree histograms via float atomics
__global__ void degree_kernel(const int* __restrict__ src, const int* __restrict__ dst,
                              float* __restrict__ deg_out, float* __restrict__ deg_in,
                              int nE) {
    for (int e = blockIdx.x * blockDim.x + threadIdx.x; e < nE;
         e += gridDim.x * blockDim.x) {
        atomicAdd(&deg_out[src[e]], 1.0f);
        atomicAdd(&deg_in[dst[e]], 1.0f);
    }
}

// 2) deg := rsqrt(max(deg,1))  (deg_out and deg_in are contiguous: one pass)
__global__ void rsqrt_kernel(float* __restrict__ deg, int n) {
    for (int i = blockIdx.x * blockDim.x + threadIdx.x; i < n;
         i += gridDim.x * blockDim.x) {
        float d = deg[i];
        d = d < 1.f ? 1.f : d;
        deg[i] = rsqrtf(d);
    }
}

// 3) edge aggregation: agg[dst] += node_feats[src] * rs_out[src]
//    one wave32 per edge, each lane owns a float4 slice (32*4 = 128 feats)
//    prefetch next edge's source row to hide the random-gather latency
__global__ __launch_bounds__(256)
void scatter_kernel(const float* __restrict__ x,
                    const int* __restrict__ src, const int* __restrict__ dst,
                    const float* __restrict__ rs_out,
                    float* __restrict__ agg, int nE) {
    const int gid    = blockIdx.x * blockDim.x + threadIdx.x;
    const int lane   = gid & 31;
    const int egrp   = gid >> 5;                       // uniform within a wave32
    const int ngrps  = (gridDim.x * blockDim.x) >> 5;
    for (int e = egrp; e < nE; e += ngrps) {
        const int s = src[e];
        const int d = dst[e];
        const int en = e + ngrps;
        if (en < nE) {   // prefetch next gather row into cache (global_prefetch_b8)
            const int sn = src[en];
            __builtin_prefetch(x + (size_t)sn * D + lane * 4, 0, 3);
        }
        const float sc = rs_out[s];
        const float4 v =
            reinterpret_cast<const float4*>(x + (size_t)s * D)[lane];
        float* ap = agg + (size_t)d * D + lane * 4;
        atomicAdd(ap + 0, v.x * sc);
        atomicAdd(ap + 1, v.y * sc);
        atomicAdd(ap + 2, v.z * sc);
        atomicAdd(ap + 3, v.w * sc);
    }
}

// 4) fused dual GEMM (WMMA f32 16x16x4) + bias + relu + add + BN partial sums
//    y = relu((agg*rs_in) @ W + b) + relu(x @ Wr + br); accumulate col sums.
//    Each wave owns (row-block, column-half): 2x4 v8f accumulators = 64 VGPRs.
//    W/Wr live in LDS pair-interleaved so each B fragment is one ds_load_b64.
__global__ __launch_bounds__(256)
void gemm_bn_partial(const float* __restrict__ x,
                     const float* __restrict__ agg,
                     const float* __restrict__ rs_in,
                     const float* __restrict__ W,  const float* __restrict__ bv,
                     const float* __restrict__ Wr, const float* __restrict__ brv,
                     float* __restrict__ y,
                     float* __restrict__ colsum, float* __restrict__ colsq) {
    // pair-interleaved: pairs[k2*D + c] = { W[2*k2][c], W[2*k2+1][c] }
    __shared__ float sWp[D * D];    // 64 KB
    __shared__ float sWrp[D * D];   // 64 KB (128 KB total, < 320 KB/WGP)
    for (int i = threadIdx.x; i < D * D; i += 256) {
        const int k   = i >> 7;          // row of W
        const int col = i & (D - 1);
        const int di  = (k >> 1) * (2 * D) + col * 2 + (k & 1);
        sWp[di]  = W[i];
        sWrp[di] = Wr[i];
    }
    __syncthreads();

    const float2* __restrict__ wpair  = reinterpret_cast<const float2*>(sWp);
    const float2* __restrict__ wrpair = reinterpret_cast<const float2*>(sWrp);

    const int lane   = threadIdx.x & 31;
    const int mhalf  = lane >> 4;        // 0/1: K/M half served by this lane
    const int mlo    = lane & 15;
    const int gwave  = (blockIdx.x * 256 + threadIdx.x) >> 5;
    const int nwaves = (gridDim.x * 256) >> 5;
    const int NTASK  = ROW_BLOCKS * 2;   // (row block, column half)

    float lsum[4], lsq[4];
#pragma unroll
    for (int t = 0; t < 4; ++t) { lsum[t] = 0.f; lsq[t] = 0.f; }

    const v8f vzero = {0.f, 0.f, 0.f, 0.f, 0.f, 0.f, 0.f, 0.f};
    int colbase = 0;  // remember last column half for the final atomics

    for (int task = gwave; task < NTASK; task += nwaves) {
        const int rb   = task >> 1;
        const int ch   = task & 1;               // column half: tiles ch*4..ch*4+3
        colbase = ch * 64;
        const int row0 = rb * 16;
        const int arow = row0 + mlo;             // A-fragment row for this lane
        const float rin = rs_in[arow];
        const float* __restrict__ xrow = x   + (size_t)arow * D;
        const float* __restrict__ grow = agg + (size_t)arow * D;

        v8f accG[4], accR[4];
#pragma unroll
        for (int t = 0; t < 4; ++t) { accG[t] = vzero; accR[t] = vzero; }

#pragma unroll 4
        for (int k0 = 0; k0 < D; k0 += 4) {
            const int ka = k0 + mhalf * 2;       // even -> aligned float2
            const float2 ax = *reinterpret_cast<const float2*>(xrow + ka);
            const float2 ag = *reinterpret_cast<const float2*>(grow + ka);
            v2f aR; aR[0] = ax.x;        aR[1] = ax.y;
            v2f aG; aG[0] = ag.x * rin;  aG[1] = ag.y * rin;
            const int k2 = (k0 >> 1) + mhalf;    // pair index
            const float2* wp  = wpair  + k2 * D;
            const float2* wrp = wrpair + k2 * D;
#pragma unroll
            for (int t = 0; t < 4; ++t) {
                const int col = colbase + t * 16 + mlo;
                const float2 g2 = wp[col];       // one ds_load_b64
                const float2 r2 = wrp[col];      // one ds_load_b64
                v2f bG; bG[0] = g2.x; bG[1] = g2.y;
                v2f bR; bR[0] = r2.x; bR[1] = r2.y;
                accG[t] = __builtin_amdgcn_wmma_f32_16x16x4_f32(
                    false, aG, false, bG, (short)0, accG[t], false, false);
                accR[t] = __builtin_amdgcn_wmma_f32_16x16x4_f32(
                    false, aR, false, bR, (short)0, accR[t], false, false);
            }
        }

        // epilogue: bias + relu + add, write y, accumulate BN partials
#pragma unroll
        for (int t = 0; t < 4; ++t) {
            const int col = colbase + t * 16 + mlo;
            const float bg = bv[col];
            const float bR = brv[col];
            float* yp = y + (size_t)(row0 + mhalf * 8) * D + col;
#pragma unroll
            for (int r = 0; r < 8; ++r) {
                float g  = accG[t][r] + bg;  g  = fmaxf(g, 0.f);
                float rr = accR[t][r] + bR;  rr = fmaxf(rr, 0.f);
                const float v = g + rr;
                yp[(size_t)r * D] = v;
                lsum[t] += v;
                lsq[t]   = fmaf(v, v, lsq[t]);
            }
        }
    }

    // NOTE: with NTASK=6250 and grid-stride, every wave keeps a single column
    // half (nwaves chosen >= NTASK or a multiple covering same parity is not
    // guaranteed, so flush partials per column-half by re-walking is needed);
    // simplest correct form: flush after each task would be atomic-heavy, so
    // we instead require nwaves >= NTASK (grid sized for one task per wave,
    // see launch below) and flush once here.
#pragma unroll
    for (int t = 0; t < 4; ++t) {
        const int col = colbase + t * 16 + mlo;
        atomicAdd(&colsum[col], lsum[t]);
        atomicAdd(&colsq[col],  lsq[t]);
    }
}

// 5) finalize BN stats -> per-column scale/shift
__global__ void bn_finalize(const float* __restrict__ colsum,
                            const float* __restrict__ colsq,
                            const float* __restrict__ gamma,
                            const float* __restrict__ beta,
                            float* __restrict__ ab) {
    const int c = threadIdx.x;
    const float inv_n = 1.0f / (float)N_NODES;
    const float mean = colsum[c] * inv_n;
    const float var  = colsq[c] * inv_n - mean * mean;
    const float a    = gamma[c] * rsqrtf(var + BN_EPS);
    ab[c]       = a;
    ab[c + 128] = beta[c] - mean * a;
}

// 6) normalize d_out in place: y = y * a[col] + c[col]
__global__ void norm_kernel(float* __restrict__ y, const float* __restrict__ ab, int n) {
    for (int i = blockIdx.x * blockDim.x + threadIdx.x; i < n;
         i += gridDim.x * blockDim.x) {
        const int c = i & (D - 1);
        y[i] = fmaf(y[i], ab[c], ab[c + 128]);
    }
}

// -------------------------------------------------------------------------
extern "C" void kernel_launch(void* const* d_in, const int* in_sizes, int n_in,
                              void* d_out, int out_size, void* d_ws, size_t ws_size,
                              hipStream_t stream) {
    const float* x     = (const float*)d_in[0];
    const int*   src   = (const int*)d_in[1];
    const int*   dst   = (const int*)d_in[2];
    const float* W     = (const float*)d_in[3];
    const float* b     = (const float*)d_in[4];
    const float* Wr    = (const float*)d_in[5];
    const float* br    = (const float*)d_in[6];
    const float* gamma = (const float*)d_in[7];
    const float* beta  = (const float*)d_in[8];

    float* ws      = (float*)d_ws;
    float* y       = (float*)d_out;
    float* deg_out = ws + DEG_OUT_OFF;
    float* deg_in  = ws + DEG_IN_OFF;
    float* colsum  = ws + COLSUM_OFF;
    float* colsq   = ws + COLSQ_OFF;
    float* ab      = ws + AB_OFF;
    float* agg     = ws + AGG_OFF;

    zero_kernel<<<1024, 256, 0, stream>>>((float4*)ws, ZERO_FLOATS / 4);
    degree_kernel<<<1024, 256, 0, stream>>>(src, dst, deg_out, deg_in, N_EDGES);
    rsqrt_kernel<<<(2 * N_NODES + 255) / 256, 256, 0, stream>>>(ws, 2 * N_NODES);
    scatter_kernel<<<2048, 256, 0, stream>>>(x, src, dst, deg_out, agg, N_EDGES);
    // one task per wave: 2*ROW_BLOCKS tasks, 8 waves per 256-thread block
    gemm_bn_partial<<<(2 * ROW_BLOCKS + 7) / 8, 256, 0, stream>>>(
        x, agg, deg_in, W, b, Wr, br, y, colsum, colsq);
    bn_finalize<<<1, 128, 0, stream>>>(colsum, colsq, gamma, beta, ab);
    norm_kernel<<<2048, 256, 0, stream>>>(y, ab, N_NODES * D);
}